// CrossNet_31198642438695
// MI455X (gfx1250) — compile-verified
//
#include <hip/hip_runtime.h>

typedef __attribute__((ext_vector_type(2))) float v2f;
typedef __attribute__((ext_vector_type(4))) float v4f;
typedef __attribute__((ext_vector_type(8))) float v8f;

#define DIM 128
#define NLAYERS 4
#define ROW_PAD 132            // LDS row stride in floats (132 % 64 == 4 -> conflict-free b64 pattern)
#define WAVES_PER_BLOCK 4
#define ROWS_PER_WAVE 16
#define THREADS (WAVES_PER_BLOCK * 32)

__global__ __launch_bounds__(THREADS) void crossnet_wmma_kernel(
    const float* __restrict__ x,        // (B, 128) row-major
    const float* __restrict__ kernels,  // (L, 128, 1) -> flat L*128
    const float* __restrict__ biases,   // (L, 128, 1) -> flat L*128
    float* __restrict__ out,            // (B, 128)
    int Brows)
{
    __shared__ float s_w[NLAYERS * DIM];
    __shared__ float s_b[NLAYERS * DIM];
    __shared__ float s_tile[WAVES_PER_BLOCK][ROWS_PER_WAVE * ROW_PAD];

    const int tid = threadIdx.x;

    // Cooperative load of all layer weights/biases (done before any wave exits).
    for (int i = tid; i < NLAYERS * DIM; i += THREADS) {
        s_w[i] = kernels[i];
        s_b[i] = biases[i];
    }
    __syncthreads();

    const int lane = tid & 31;
    const int r    = lane & 15;   // tile row owned by this lane
    const int h    = lane >> 4;   // half: K-offset = 2h within each group of 4
    // Uniform (scalar) wave id so the OOB branch is a scalar branch (WMMA needs EXEC==all-ones).
    const int wave = __builtin_amdgcn_readfirstlane(tid >> 5);
    const int r0   = (blockIdx.x * WAVES_PER_BLOCK + wave) * ROWS_PER_WAVE;
    if (r0 >= Brows) return;      // B % 16 == 0 -> tiles are all-in or all-out

    float* tile = &s_tile[wave][0];

    // ---- Stage x0 tile into LDS, fully coalesced: 512B contiguous per wave per iter ----
    #pragma unroll
    for (int t = 0; t < ROWS_PER_WAVE; ++t) {
        v4f v = *(const v4f*)(x + (size_t)(r0 + t) * DIM + 4 * lane);
        *(v4f*)(tile + t * ROW_PAD + 4 * lane) = v;
    }
    asm volatile("s_wait_dscnt 0x0" ::: "memory");

    // ---- Pull xl into registers in the B-matrix (row-per-lane) layout ----
    // lane holds row r, columns 4s + 2h + {0,1}; ds_load_b64 bank-conflict-free via ROW_PAD.
    const float* myrow = tile + r * ROW_PAD + 2 * h;
    float xl[64];
    #pragma unroll
    for (int s = 0; s < 32; ++s) {
        v2f p = *(const v2f*)(myrow + 4 * s);
        xl[2 * s]     = p.x;
        xl[2 * s + 1] = p.y;
    }

    // ---- 4 cross layers, all in registers ----
    #pragma unroll
    for (int l = 0; l < NLAYERS; ++l) {
        const float* wl = s_w + l * DIM + 2 * h;
        const float* bl = s_b + l * DIM + 2 * h;

        // dot[n] = sum_d xl[n][d] * w[d] via f32 WMMA:
        //   A[m][k] = w[4s+k]  (broadcast over M; same per-lane layout as B)
        //   B[k][n] = xl[n][4s+k]
        //   => D[m][n] = dot[n], broadcast over m -> every accumulator elem of
        //      lane l holds dot[l%16] == dot for the row this lane owns.
        v8f c = {};
        #pragma unroll
        for (int s = 0; s < 32; ++s) {
            v2f a = *(const v2f*)(wl + 4 * s);           // w broadcast (LDS, broadcast read)
            v2f b = { xl[2 * s], xl[2 * s + 1] };        // xl operand
            c = __builtin_amdgcn_wmma_f32_16x16x4_f32(
                    /*neg_a=*/false, a, /*neg_b=*/false, b,
                    /*c_mod=*/(short)0, c, /*reuse_a=*/false, /*reuse_b=*/false);
        }
        const float val = c[0];

        // xl = x0 * val + b + xl   (x0 re-read from the LDS tile)
        #pragma unroll
        for (int s = 0; s < 32; ++s) {
            v2f x0p = *(const v2f*)(myrow + 4 * s);
            v2f bp  = *(const v2f*)(bl + 4 * s);
            xl[2 * s]     = fmaf(x0p.x, val, bp.x + xl[2 * s]);
            xl[2 * s + 1] = fmaf(x0p.y, val, bp.y + xl[2 * s + 1]);
        }
    }

    // ---- Stage result back through LDS for coalesced stores ----
    #pragma unroll
    for (int s = 0; s < 32; ++s) {
        v2f p = { xl[2 * s], xl[2 * s + 1] };
        *(v2f*)(tile + r * ROW_PAD + 2 * h + 4 * s) = p;
    }
    asm volatile("s_wait_dscnt 0x0" ::: "memory");

    #pragma unroll
    for (int t = 0; t < ROWS_PER_WAVE; ++t) {
        v4f v = *(const v4f*)(tile + t * ROW_PAD + 4 * lane);
        *(v4f*)(out + (size_t)(r0 + t) * DIM + 4 * lane) = v;
    }
}

extern "C" void kernel_launch(void* const* d_in, const int* in_sizes, int n_in,
                              void* d_out, int out_size, void* d_ws, size_t ws_size,
                              hipStream_t stream) {
    const float* x       = (const float*)d_in[0];  // inputs  (B, 128) f32
    const float* kernels = (const float*)d_in[1];  // (4, 128, 1) f32
    const float* biases  = (const float*)d_in[2];  // (4, 128, 1) f32
    float* out = (float*)d_out;

    const int Brows  = in_sizes[0] / DIM;                       // 500000
    const int rowsPerBlock = WAVES_PER_BLOCK * ROWS_PER_WAVE;   // 64
    const int blocks = (Brows + rowsPerBlock - 1) / rowsPerBlock;

    crossnet_wmma_kernel<<<blocks, THREADS, 0, stream>>>(x, kernels, biases, out, Brows);
}